// GINConv_57672820851271
// MI455X (gfx1250) — compile-verified
//
#include <hip/hip_runtime.h>
#include <hip/hip_bf16.h>

typedef __attribute__((ext_vector_type(2))) float v2f;
typedef __attribute__((ext_vector_type(8))) float v8f;

#define N_NODES 100000
#define N_EDGES 640000
#define DIM 128
#define SA 132                   // padded LDS row stride (floats): 16B-aligned rows, conflict-free frags
#define N_TILES (N_NODES / 16)   // 6250
#define MLP_BLOCKS 320

// ---------------- Kernel 1: agg = (1 + eps) * x  (residual pre-fused) ----------------
__global__ __launch_bounds__(256) void gin_init_kernel(const float* __restrict__ x,
                                                       const float* __restrict__ eps,
                                                       float* __restrict__ agg, int n4) {
    int i = blockIdx.x * blockDim.x + threadIdx.x;
    if (i >= n4) return;
    float s = 1.0f + eps[0];
    float4 v = ((const float4*)x)[i];
    v.x *= s; v.y *= s; v.z *= s; v.w *= s;
    ((float4*)agg)[i] = v;
}

// ---------------- Kernel 2: scatter-add; LDS-staged indices, one wave per edge ----------------
__global__ __launch_bounds__(256) void gin_scatter_kernel(const float* __restrict__ x,
                                                          const int* __restrict__ eidx,
                                                          float* __restrict__ agg) {
    __shared__ int s_src[64];
    __shared__ int s_dst[64];
    const int tid = threadIdx.x;
    const int ebase = blockIdx.x * 64;          // N_EDGES == 64 * gridDim.x exactly, no tail
    if (tid < 64)       s_src[tid]      = eidx[ebase + tid];
    else if (tid < 128) s_dst[tid - 64] = eidx[N_EDGES + ebase + (tid - 64)];
    __syncthreads();

    const int wave = tid >> 5;
    const int lane = tid & 31;
    #pragma unroll
    for (int j = 0; j < 8; ++j) {
        int le = wave * 8 + j;
        const float* s = x   + (size_t)s_src[le] * DIM + lane * 4;
        float*       d = agg + (size_t)s_dst[le] * DIM + lane * 4;
        float4 v = *(const float4*)s;
        atomicAdd(d + 0, v.x);
        atomicAdd(d + 1, v.y);
        atomicAdd(d + 2, v.z);
        atomicAdd(d + 3, v.w);
    }
}

// Issue one 16x128 f32 tile (row-major, LDS row stride SA) as 16 async b128 row copies.
__device__ __forceinline__ void async_stage_tile(const float* __restrict__ gsrc,
                                                 float* lbuf, int lane) {
    unsigned lbase = (unsigned)(size_t)lbuf + (unsigned)lane * 16u;
    const char* g = (const char*)gsrc + lane * 16;
    #pragma unroll
    for (int it = 0; it < 16; ++it) {
        unsigned laddr = lbase + (unsigned)(it * SA * 4);
        unsigned long long gaddr = (unsigned long long)(g + it * 512);
        asm volatile("global_load_async_to_lds_b128 %0, %1, off"
                     :: "v"(laddr), "v"(gaddr) : "memory");
    }
}

__device__ __forceinline__ void wait_async_zero() {
    asm volatile("s_wait_asynccnt 0" ::: "memory");
}

// ---------------- Kernel 3: fused MLP  out = relu(agg@W1+b1)@W2+b2 via f32 WMMA ----------------
__global__ __launch_bounds__(256) void gin_mlp_kernel(const float* __restrict__ agg,
                                                      const float* __restrict__ W1,
                                                      const float* __restrict__ b1,
                                                      const float* __restrict__ W2,
                                                      const float* __restrict__ b2,
                                                      float* __restrict__ out) {
    __shared__ float w1t[DIM * SA];        // w1t[c*SA + r] = W1[r][c]
    __shared__ float w2t[DIM * SA];
    __shared__ float bias1[DIM];
    __shared__ float bias2[DIM];
    __shared__ float atile[8][2][16 * SA]; // per-wave double-buffered A / h tile

    const int tid = threadIdx.x;

    // Cooperative transposed fill of both weight matrices (one-time per block).
    for (int idx = tid; idx < DIM * DIM; idx += 256) {
        int r = idx >> 7;
        int c = idx & 127;
        w1t[c * SA + r] = W1[idx];
        w2t[c * SA + r] = W2[idx];
    }
    if (tid < DIM) { bias1[tid] = b1[tid]; bias2[tid] = b2[tid]; }
    __syncthreads();

    const int wave = tid >> 5;
    const int lane = tid & 31;
    const int half = lane >> 4;   // selects K pair {0,1} vs {2,3} within each k-step
    const int l15  = lane & 15;   // M (A) / N (B,C,D) index

    const int stride = gridDim.x * 8;
    int tile = blockIdx.x * 8 + wave;
    int buf = 0;

    if (tile < N_TILES)
        async_stage_tile(agg + (size_t)tile * 16 * DIM, atile[wave][0], lane);

    for (; tile < N_TILES; tile += stride, buf ^= 1) {
        wait_async_zero();                 // current buffer's tile has landed in LDS
        int nxt = tile + stride;
        if (nxt < N_TILES)                 // prefetch next tile while we compute
            async_stage_tile(agg + (size_t)nxt * 16 * DIM, atile[wave][buf ^ 1], lane);

        float* at = atile[wave][buf];
        v8f acc[8];

        // ---- layer 1: acc[n] = A(16x128) @ W1(:,16n..16n+15) + b1 ----
        #pragma unroll
        for (int n = 0; n < 8; ++n) {
            float bv = bias1[n * 16 + l15];
            #pragma unroll
            for (int r = 0; r < 8; ++r) acc[n][r] = bv;
        }
        for (int kk = 0; kk < 32; ++kk) {
            int k0 = kk * 4 + half * 2;
            v2f a = *(const v2f*)&at[l15 * SA + k0];
            #pragma unroll
            for (int n = 0; n < 8; ++n) {
                v2f b = *(const v2f*)&w1t[(n * 16 + l15) * SA + k0];
                acc[n] = __builtin_amdgcn_wmma_f32_16x16x4_f32(
                    false, a, false, b, (short)0, acc[n], false, false);
            }
        }

        // ---- ReLU, bounce h through the same buffer (C layout -> row major) ----
        #pragma unroll
        for (int n = 0; n < 8; ++n) {
            #pragma unroll
            for (int r = 0; r < 8; ++r) {
                int row = r + half * 8;
                float v = acc[n][r];
                at[row * SA + n * 16 + l15] = (v > 0.0f) ? v : 0.0f;
            }
        }

        // ---- layer 2: acc[n] = h(16x128) @ W2(:,16n..) + b2 ----
        #pragma unroll
        for (int n = 0; n < 8; ++n) {
            float bv = bias2[n * 16 + l15];
            #pragma unroll
            for (int r = 0; r < 8; ++r) acc[n][r] = bv;
        }
        for (int kk = 0; kk < 32; ++kk) {
            int k0 = kk * 4 + half * 2;
            v2f a = *(const v2f*)&at[l15 * SA + k0];
            #pragma unroll
            for (int n = 0; n < 8; ++n) {
                v2f b = *(const v2f*)&w2t[(n * 16 + l15) * SA + k0];
                acc[n] = __builtin_amdgcn_wmma_f32_16x16x4_f32(
                    false, a, false, b, (short)0, acc[n], false, false);
            }
        }

        // ---- store 16x128 output tile ----
        float* dst = out + (size_t)tile * 16 * DIM;
        #pragma unroll
        for (int n = 0; n < 8; ++n) {
            #pragma unroll
            for (int r = 0; r < 8; ++r) {
                int row = r + half * 8;
                dst[row * DIM + n * 16 + l15] = acc[n][r];
            }
        }
    }
}

extern "C" void kernel_launch(void* const* d_in, const int* in_sizes, int n_in,
                              void* d_out, int out_size, void* d_ws, size_t ws_size,
                              hipStream_t stream) {
    const float* x    = (const float*)d_in[0];
    const int*   eidx = (const int*)d_in[1];
    const float* eps  = (const float*)d_in[2];
    const float* W1   = (const float*)d_in[3];
    const float* b1   = (const float*)d_in[4];
    const float* W2   = (const float*)d_in[5];
    const float* b2   = (const float*)d_in[6];
    float* out = (float*)d_out;
    float* agg = (float*)d_ws;   // N_NODES*DIM floats = 51.2 MB scratch

    // 1) agg = (1+eps)*x
    int n4 = N_NODES * DIM / 4;
    gin_init_kernel<<<(n4 + 255) / 256, 256, 0, stream>>>(x, eps, agg, n4);

    // 2) agg[dst] += x[src]  (64 edges per block, one wave per edge)
    gin_scatter_kernel<<<N_EDGES / 64, 256, 0, stream>>>(x, eidx, agg);

    // 3) fused MLP with f32 WMMA + async double-buffered tile staging
    gin_mlp_kernel<<<MLP_BLOCKS, 256, 0, stream>>>(agg, W1, b1, W2, b2, out);
}